// Skeleton_54829552501239
// MI455X (gfx1250) — compile-verified
//
#include <hip/hip_runtime.h>

#ifndef __has_builtin
#define __has_builtin(x) 0
#endif

#define NJ 24
#define BLK 128
// LDS layout (float units). Strides padded: 148 = 144+4 (16B-aligned rows,
// <=2-way bank conflict), 76 = 72+4.
#define IN_STRIDE_F 148
#define OUT_STRIDE_F 76
#define L_XYZ 0
#define L_IN 72
#define L_OUT (L_IN + BLK * IN_STRIDE_F)
#define LDS_FLOATS (L_OUT + BLK * OUT_STRIDE_F)

#if __has_builtin(__builtin_amdgcn_global_load_async_to_lds_b128) && \
    __has_builtin(__builtin_amdgcn_global_store_async_from_lds_b128)
#define HAVE_ASYNC 1
#else
#define HAVE_ASYNC 0
#endif

typedef int v4i __attribute__((vector_size(16)));
typedef __attribute__((address_space(1))) v4i* gp_v4i;
typedef __attribute__((address_space(3))) v4i* lp_v4i;

__device__ __forceinline__ void g2l_b128(void* g, void* l) {
#if HAVE_ASYNC
  __builtin_amdgcn_global_load_async_to_lds_b128((gp_v4i)g, (lp_v4i)l, 0, 0);
#else
  *(float4*)l = *(const float4*)g;
#endif
}

__device__ __forceinline__ void l2g_b128(void* g, void* l) {
#if HAVE_ASYNC
  __builtin_amdgcn_global_store_async_from_lds_b128((gp_v4i)g, (lp_v4i)l, 0, 0);
#else
  *(float4*)g = *(const float4*)l;
#endif
}

__device__ __forceinline__ void wait_async() {
#if HAVE_ASYNC
#if __has_builtin(__builtin_amdgcn_s_wait_asynccnt)
  __builtin_amdgcn_s_wait_asynccnt(0);
#else
  asm volatile("s_wait_asynccnt 0" ::: "memory");
#endif
#endif
}

__device__ __forceinline__ float fast_rsqrt(float x) {
#if __has_builtin(__builtin_amdgcn_rsqf)
  return __builtin_amdgcn_rsqf(x);   // single v_rsq_f32 (~1 ulp)
#else
  return rsqrtf(x);
#endif
}

// Zhou et al. 6D -> 3x3 rotation (rows b1,b2,b3), all in registers.
__device__ __forceinline__ void rot6d(const float* __restrict__ d6, float R[9]) {
  float a0 = d6[0], a1 = d6[1], a2 = d6[2];
  float b0 = d6[3], b1 = d6[4], b2 = d6[5];
  float n1 = fast_rsqrt(fmaf(a0, a0, fmaf(a1, a1, a2 * a2)));
  float r0 = a0 * n1, r1 = a1 * n1, r2 = a2 * n1;
  float d = fmaf(r0, b0, fmaf(r1, b1, r2 * b2));
  float c0 = fmaf(-d, r0, b0), c1 = fmaf(-d, r1, b1), c2 = fmaf(-d, r2, b2);
  float n2 = fast_rsqrt(fmaf(c0, c0, fmaf(c1, c1, c2 * c2)));
  c0 *= n2; c1 *= n2; c2 *= n2;
  R[0] = r0; R[1] = r1; R[2] = r2;
  R[3] = c0; R[4] = c1; R[5] = c2;
  R[6] = r1 * c2 - r2 * c1;          // b3 = b1 x b2
  R[7] = r2 * c0 - r0 * c2;
  R[8] = r0 * c1 - r1 * c0;
}

// One joint: Rg = Rp @ rot(j); tg = Rp @ (xyz[j]-xyz[p]) + tp; emit tg to LDS.
// Safe when (Rg,tg) alias (Rp,tp).
__device__ __forceinline__ void joint_step(const float* __restrict__ A,
                                           const float* __restrict__ X,
                                           float* __restrict__ O,
                                           int j, int p,
                                           const float* Rp, const float* tp,
                                           float* Rg, float* tg) {
  float R[9];
  rot6d(A + 6 * j, R);
  float ox = X[3 * j + 0] - X[3 * p + 0];
  float oy = X[3 * j + 1] - X[3 * p + 1];
  float oz = X[3 * j + 2] - X[3 * p + 2];
  float nt0 = fmaf(Rp[0], ox, fmaf(Rp[1], oy, fmaf(Rp[2], oz, tp[0])));
  float nt1 = fmaf(Rp[3], ox, fmaf(Rp[4], oy, fmaf(Rp[5], oz, tp[1])));
  float nt2 = fmaf(Rp[6], ox, fmaf(Rp[7], oy, fmaf(Rp[8], oz, tp[2])));
  float T[9];
#pragma unroll
  for (int r = 0; r < 3; ++r)
#pragma unroll
    for (int k = 0; k < 3; ++k)
      T[r * 3 + k] = fmaf(Rp[r * 3 + 0], R[k],
                     fmaf(Rp[r * 3 + 1], R[3 + k], Rp[r * 3 + 2] * R[6 + k]));
#pragma unroll
  for (int q = 0; q < 9; ++q) Rg[q] = T[q];
  tg[0] = nt0; tg[1] = nt1; tg[2] = nt2;
  O[3 * j + 0] = nt0; O[3 * j + 1] = nt1; O[3 * j + 2] = nt2;
}

__global__ __launch_bounds__(BLK) void fk_kernel(const float* __restrict__ angles,
                                                 const float* __restrict__ xyz,
                                                 float* __restrict__ out,
                                                 int batch) {
  extern __shared__ float smem[];
  const unsigned tid = threadIdx.x;
  const long long base = (long long)blockIdx.x * BLK;
  const int rem = (int)(batch - base);       // elements this block owns (>0)

  if (tid < 72) smem[L_XYZ + tid] = xyz[tid];

  // ---- Stage angles: coalesced global reads, scattered (padded) LDS writes.
  {
    const float* gin = angles + base * (NJ * 6);
    if (rem >= BLK) {                        // full block: branch-free clause
#pragma unroll
      for (unsigned i = 0; i < 36; ++i) {    // 128 elems * 36 float4 chunks
        unsigned c = tid + i * BLK;
        unsigned e = c / 36u;
        unsigned p = c - e * 36u;
        g2l_b128((void*)(gin + e * 144u + p * 4u),
                 &smem[L_IN + e * IN_STRIDE_F + p * 4]);
      }
    } else {                                 // ragged tail block
#pragma unroll 4
      for (unsigned i = 0; i < 36; ++i) {
        unsigned c = tid + i * BLK;
        unsigned e = c / 36u;
        unsigned p = c - e * 36u;
        if ((int)e < rem)
          g2l_b128((void*)(gin + e * 144u + p * 4u),
                   &smem[L_IN + e * IN_STRIDE_F + p * 4]);
      }
    }
  }
  wait_async();
  __syncthreads();

  // ---- Per-lane forward kinematics (DFS over the tree; only root and
  // joint-9 transforms stay live across branches).
  {
    const float* A = &smem[L_IN + tid * IN_STRIDE_F];
    float* O = &smem[L_OUT + tid * OUT_STRIDE_F];
    const float* X = &smem[L_XYZ];

    float R0[9], t0[3], Rc[9], tc[3], R9[9], t9[3];
    rot6d(A, R0);
    t0[0] = X[0]; t0[1] = X[1]; t0[2] = X[2];
    O[0] = t0[0]; O[1] = t0[1]; O[2] = t0[2];

    // branch: 1 -> 4 -> 7 -> 10
    joint_step(A, X, O, 1, 0, R0, t0, Rc, tc);
    joint_step(A, X, O, 4, 1, Rc, tc, Rc, tc);
    joint_step(A, X, O, 7, 4, Rc, tc, Rc, tc);
    joint_step(A, X, O, 10, 7, Rc, tc, Rc, tc);
    // branch: 2 -> 5 -> 8 -> 11
    joint_step(A, X, O, 2, 0, R0, t0, Rc, tc);
    joint_step(A, X, O, 5, 2, Rc, tc, Rc, tc);
    joint_step(A, X, O, 8, 5, Rc, tc, Rc, tc);
    joint_step(A, X, O, 11, 8, Rc, tc, Rc, tc);
    // spine: 3 -> 6 -> 9
    joint_step(A, X, O, 3, 0, R0, t0, Rc, tc);
    joint_step(A, X, O, 6, 3, Rc, tc, Rc, tc);
    joint_step(A, X, O, 9, 6, Rc, tc, R9, t9);
    // 12 -> 15
    joint_step(A, X, O, 12, 9, R9, t9, Rc, tc);
    joint_step(A, X, O, 15, 12, Rc, tc, Rc, tc);
    // 13 -> 16 -> 18 -> 20 -> 22
    joint_step(A, X, O, 13, 9, R9, t9, Rc, tc);
    joint_step(A, X, O, 16, 13, Rc, tc, Rc, tc);
    joint_step(A, X, O, 18, 16, Rc, tc, Rc, tc);
    joint_step(A, X, O, 20, 18, Rc, tc, Rc, tc);
    joint_step(A, X, O, 22, 20, Rc, tc, Rc, tc);
    // 14 -> 17 -> 19 -> 21 -> 23
    joint_step(A, X, O, 14, 9, R9, t9, Rc, tc);
    joint_step(A, X, O, 17, 14, Rc, tc, Rc, tc);
    joint_step(A, X, O, 19, 17, Rc, tc, Rc, tc);
    joint_step(A, X, O, 21, 19, Rc, tc, Rc, tc);
    joint_step(A, X, O, 23, 21, Rc, tc, Rc, tc);
  }

  __syncthreads();

  // ---- Drain results: padded LDS rows -> coalesced global stores.
  {
    float* gout = out + base * (NJ * 3);
    if (rem >= BLK) {
#pragma unroll
      for (unsigned i = 0; i < 18; ++i) {    // 128 elems * 18 float4 chunks
        unsigned c = tid + i * BLK;
        unsigned e = c / 18u;
        unsigned p = c - e * 18u;
        l2g_b128((void*)(gout + e * 72u + p * 4u),
                 &smem[L_OUT + e * OUT_STRIDE_F + p * 4]);
      }
    } else {
#pragma unroll 4
      for (unsigned i = 0; i < 18; ++i) {
        unsigned c = tid + i * BLK;
        unsigned e = c / 18u;
        unsigned p = c - e * 18u;
        if ((int)e < rem)
          l2g_b128((void*)(gout + e * 72u + p * 4u),
                   &smem[L_OUT + e * OUT_STRIDE_F + p * 4]);
      }
    }
  }
  wait_async();
}

extern "C" void kernel_launch(void* const* d_in, const int* in_sizes, int n_in,
                              void* d_out, int out_size, void* d_ws, size_t ws_size,
                              hipStream_t stream) {
  const float* angles = (const float*)d_in[0];
  const float* xyz = (const float*)d_in[1];
  float* out = (float*)d_out;
  int batch = in_sizes[0] / (NJ * 6);
  int grid = (batch + BLK - 1) / BLK;
  size_t shmem = (size_t)LDS_FLOATS * sizeof(float);
  fk_kernel<<<grid, BLK, shmem, stream>>>(angles, xyz, out, batch);
}